// LaplacianTemplateLoss_78615081386503
// MI455X (gfx1250) — compile-verified
//
#include <hip/hip_runtime.h>
#include <stdint.h>

// Problem constants (from the reference): B=16, N=100000, K=8, scalar output.
#define BATCH   16
#define NPTS    100000
#define KNB     8
#define TILE_N  256
#define THREADS 256
#define NBLOCKS ((NPTS + TILE_N - 1) / TILE_N)   // 391
#define NWAVES  (THREADS / 32)

// ---------------------------------------------------------------------------
// CDNA5 helpers
// ---------------------------------------------------------------------------

// Flat pointer into LDS: hardware takes addr[31:0] as the LDS byte address.
__device__ __forceinline__ uint32_t lds_offset(const void* p) {
  return (uint32_t)(uintptr_t)p;
}

// GVS-form async copy global -> LDS, 16 bytes per lane. Tracked by ASYNCcnt.
__device__ __forceinline__ void async_load_b128(uint32_t lds_addr, uint32_t voff,
                                                const void* sbase) {
  asm volatile("global_load_async_to_lds_b128 %0, %1, %2"
               :
               : "v"(lds_addr), "v"(voff), "s"(sbase)
               : "memory");
}

__device__ __forceinline__ void wait_async_all() {
  asm volatile("s_wait_asynccnt 0" ::: "memory");
}

// wave32 butterfly reduction
__device__ __forceinline__ float wave_reduce(float v) {
#pragma unroll
  for (int off = 16; off > 0; off >>= 1) v += __shfl_xor(v, off, 32);
  return v;
}

// ---------------------------------------------------------------------------
// Pass 1: d = geom - geom_template  (pure streaming, float4-vectorized)
// ---------------------------------------------------------------------------
__global__ void lap_diff_kernel(const float4* __restrict__ g,
                                const float4* __restrict__ t,
                                float4* __restrict__ d, int n4) {
  int i = blockIdx.x * blockDim.x + threadIdx.x;
  if (i < n4) {
    float4 a = g[i];
    float4 b = t[i];
    d[i] = make_float4(a.x - b.x, a.y - b.y, a.z - b.z, a.w - b.w);
  }
}

// ---------------------------------------------------------------------------
// Pass 2: per-tile weighted-gather + squared-sum partial reduction.
//   USE_D  : A = precomputed difference array, Bm unused (half gather traffic)
//   ATOMIC : accumulate straight into *partials (fallback when ws is tiny)
// ---------------------------------------------------------------------------
template <bool USE_D, bool ATOMIC>
__global__ void lap_main_kernel(const float* __restrict__ A,
                                const float* __restrict__ Bm,
                                const int* __restrict__ nbs_idx,
                                const float* __restrict__ nbs_w,
                                float* __restrict__ partials) {
  __shared__ int   s_idx[TILE_N * KNB];   // 8 KB
  __shared__ float s_w[TILE_N * KNB];     // 8 KB
  __shared__ float s_part[NWAVES];

  const int n0 = blockIdx.x * TILE_N;
  const int t  = threadIdx.x;

  // ---- Stage the tile's idx/weights into LDS with async copies.
  // Tile = TILE_N*KNB = 2048 elements per array; 16B per lane per op:
  // 2 ops per thread per array cover it. Clamp tail-tile OOB lanes to 0
  // (their data is never consumed: those lanes map to n >= NPTS).
  {
    const uint32_t totBytes = (uint32_t)NPTS * KNB * 4u;
    uint32_t b0 = (uint32_t)(n0 * KNB + t * 4) * 4u;
    uint32_t b1 = b0 + 4096u;
    if (b0 + 16u > totBytes) b0 = 0u;
    if (b1 + 16u > totBytes) b1 = 0u;
    const uint32_t li = lds_offset(s_idx) + (uint32_t)t * 16u;
    const uint32_t lw = lds_offset(s_w)   + (uint32_t)t * 16u;
    async_load_b128(li,         b0, nbs_idx);
    async_load_b128(li + 4096u, b1, nbs_idx);
    async_load_b128(lw,         b0, nbs_w);
    async_load_b128(lw + 4096u, b1, nbs_w);
    wait_async_all();
  }
  __syncthreads();

  const int n = n0 + t;
  float ssq = 0.0f;

  if (n < NPTS) {
    int   ji[KNB];
    float wi[KNB];
#pragma unroll
    for (int k = 0; k < KNB; ++k) {
      ji[k] = 3 * s_idx[t * KNB + k];
      wi[k] = s_w[t * KNB + k];
    }
    const int nc = 3 * n;

    // Reuse the staged idx/w for all 16 batches.
    for (int b = 0; b < BATCH; ++b) {
      const float* __restrict__ Ab = A + (size_t)b * (NPTS * 3);
      float dx, dy, dz;
      if (USE_D) {
        dx = Ab[nc + 0];
        dy = Ab[nc + 1];
        dz = Ab[nc + 2];
#pragma unroll
        for (int k = 0; k < KNB; ++k) {
          const float* __restrict__ p = Ab + ji[k];
          dx = fmaf(wi[k], p[0], dx);
          dy = fmaf(wi[k], p[1], dy);
          dz = fmaf(wi[k], p[2], dz);
        }
      } else {
        const float* __restrict__ Bb = Bm + (size_t)b * (NPTS * 3);
        dx = Ab[nc + 0] - Bb[nc + 0];
        dy = Ab[nc + 1] - Bb[nc + 1];
        dz = Ab[nc + 2] - Bb[nc + 2];
#pragma unroll
        for (int k = 0; k < KNB; ++k) {
          const float* __restrict__ pa = Ab + ji[k];
          const float* __restrict__ pb = Bb + ji[k];
          dx = fmaf(wi[k], pa[0] - pb[0], dx);
          dy = fmaf(wi[k], pa[1] - pb[1], dy);
          dz = fmaf(wi[k], pa[2] - pb[2], dz);
        }
      }
      ssq = fmaf(dx, dx, fmaf(dy, dy, fmaf(dz, dz, ssq)));
    }
  }

  // ---- block reduction: wave32 shuffle, then cross-wave via LDS
  ssq = wave_reduce(ssq);
  if ((t & 31) == 0) s_part[t >> 5] = ssq;
  __syncthreads();
  if (t == 0) {
    float s = 0.0f;
#pragma unroll
    for (int i = 0; i < NWAVES; ++i) s += s_part[i];
    if (ATOMIC) {
      atomicAdd(partials, s);
    } else {
      partials[blockIdx.x] = s;
    }
  }
}

// ---------------------------------------------------------------------------
// Pass 3: deterministic final reduction of per-block partials -> scalar mean
// ---------------------------------------------------------------------------
__global__ void lap_reduce_kernel(const float* __restrict__ partials, int n,
                                  float* __restrict__ out, float scale) {
  __shared__ float s_part[NWAVES];
  float s = 0.0f;
  for (int i = threadIdx.x; i < n; i += blockDim.x) s += partials[i];
  s = wave_reduce(s);
  if ((threadIdx.x & 31) == 0) s_part[threadIdx.x >> 5] = s;
  __syncthreads();
  if (threadIdx.x == 0) {
    float tot = 0.0f;
#pragma unroll
    for (int i = 0; i < NWAVES; ++i) tot += s_part[i];
    out[0] = tot * scale;
  }
}

__global__ void lap_zero_kernel(float* __restrict__ out) { out[0] = 0.0f; }
__global__ void lap_scale_kernel(float* __restrict__ out, float scale) { out[0] *= scale; }

// ---------------------------------------------------------------------------
// Host launcher
// ---------------------------------------------------------------------------
extern "C" void kernel_launch(void* const* d_in, const int* in_sizes, int n_in,
                              void* d_out, int out_size, void* d_ws, size_t ws_size,
                              hipStream_t stream) {
  (void)in_sizes; (void)n_in; (void)out_size;

  const float* geom = (const float*)d_in[0];
  const float* tmpl = (const float*)d_in[1];
  const int*   idx  = (const int*)d_in[2];
  const float* w    = (const float*)d_in[3];
  float* out = (float*)d_out;

  const size_t dElems    = (size_t)BATCH * NPTS * 3;         // 4.8M floats
  const size_t dBytes    = dElems * sizeof(float);           // 19.2 MB
  const size_t partBytes = (size_t)NBLOCKS * sizeof(float);  // ~1.6 KB
  const float  scale     = 1.0f / (float)dElems;

  if (ws_size >= dBytes + partBytes) {
    // Optimal path: precompute d (halves gather traffic), then gather from d.
    float* dbuf = (float*)d_ws;
    float* part = (float*)((char*)d_ws + dBytes);
    const int n4 = (int)(dElems / 4);
    lap_diff_kernel<<<(n4 + THREADS - 1) / THREADS, THREADS, 0, stream>>>(
        (const float4*)geom, (const float4*)tmpl, (float4*)dbuf, n4);
    lap_main_kernel<true, false><<<NBLOCKS, THREADS, 0, stream>>>(
        dbuf, nullptr, idx, w, part);
    lap_reduce_kernel<<<1, THREADS, 0, stream>>>(part, NBLOCKS, out, scale);
  } else if (ws_size >= partBytes) {
    // Fused path: gather from both arrays, deterministic 2-stage reduce.
    float* part = (float*)d_ws;
    lap_main_kernel<false, false><<<NBLOCKS, THREADS, 0, stream>>>(
        geom, tmpl, idx, w, part);
    lap_reduce_kernel<<<1, THREADS, 0, stream>>>(part, NBLOCKS, out, scale);
  } else {
    // Minimal-workspace fallback: atomic accumulation into d_out, then scale.
    lap_zero_kernel<<<1, 1, 0, stream>>>(out);
    lap_main_kernel<false, true><<<NBLOCKS, THREADS, 0, stream>>>(
        geom, tmpl, idx, w, out);
    lap_scale_kernel<<<1, 1, 0, stream>>>(out, scale);
  }
}